// GPRGNN_39075612459326
// MI455X (gfx1250) — compile-verified
//
#include <hip/hip_runtime.h>
#include <stdint.h>

#define F_DIM 64
#define SAMPLING_RATE 0.3f
#define TAU_INV (1.0f / 0.3f)
#define EPS_REF 1e-8f

typedef __attribute__((ext_vector_type(2))) float v2f;
typedef __attribute__((ext_vector_type(4))) float v4f;
typedef __attribute__((ext_vector_type(8))) float v8f;

// Hardware f32 atomic add (global_atomic_add_f32); no CAS loop.
__device__ __forceinline__ void atomAddF(float* p, float v) {
#if defined(__HIP_DEVICE_COMPILE__)
  unsafeAtomicAdd(p, v);
#else
  atomicAdd(p, v);
#endif
}

__device__ __forceinline__ float pick8(v8f c, int v) {
  float r = c[0];
  r = (v == 1) ? c[1] : r;
  r = (v == 2) ? c[2] : r;
  r = (v == 3) ? c[3] : r;
  r = (v == 4) ? c[4] : r;
  r = (v == 5) ? c[5] : r;
  r = (v == 6) ? c[6] : r;
  r = (v == 7) ? c[7] : r;
  return r;
}

__device__ __forceinline__ float splitmix_u01(uint64_t s) {
  s += 0x9E3779B97F4A7C15ULL;
  s = (s ^ (s >> 30)) * 0xBF58476D1CE4E5B9ULL;
  s = (s ^ (s >> 27)) * 0x94D049BB133111EBULL;
  s ^= s >> 31;
  uint32_t m = (uint32_t)(s >> 40);  // 24 random bits
  return ((float)m + 0.5f) * (1.0f / 16777216.0f);  // in (0,1)
}

// ---------------- graph-level prep ----------------

__global__ void k_pred(const float* __restrict__ cls, int* __restrict__ pred,
                       int* __restrict__ cs, int* __restrict__ cd, int N, int C) {
  int n = blockIdx.x * blockDim.x + threadIdx.x;
  if (n >= N) return;
  const float* row = cls + (size_t)n * C;
  int best = 0;
  float bv = row[0];
  for (int c = 1; c < C; ++c) {
    float v = row[c];
    if (v > bv) { bv = v; best = c; }
  }
  pred[n] = best;
  cs[n] = 0;
  cd[n] = 0;
}

__global__ void k_init_x(const float* __restrict__ x, const float* __restrict__ temp,
                         float* __restrict__ xcur, float* __restrict__ hidden, long long total4) {
  long long i = (long long)blockIdx.x * blockDim.x + threadIdx.x;
  if (i >= total4) return;
  v4f v = *(const v4f*)(x + i * 4);
  *(v4f*)(xcur + i * 4) = v;
  float t0 = temp[0];
  v4f h = {t0 * v[0], t0 * v[1], t0 * v[2], t0 * v[3]};
  *(v4f*)(hidden + i * 4) = h;
}

__global__ void k_homcnt(const int* __restrict__ src, const int* __restrict__ dst,
                         const int* __restrict__ pred, int* __restrict__ cs,
                         int* __restrict__ cd, int E) {
  int e = blockIdx.x * blockDim.x + threadIdx.x;
  if (e >= E) return;
  int s = src[e], d = dst[e];
  if (s == d) return;  // nonself mask
  atomicAdd(&cd[s], 1);
  if (pred[s] == pred[d]) atomicAdd(&cs[s], 1);
}

__global__ void k_homfin(const int* __restrict__ cs, const int* __restrict__ cd,
                         float* __restrict__ hom, int N) {
  int n = blockIdx.x * blockDim.x + threadIdx.x;
  if (n >= N) return;
  int d = cd[n];
  hom[n] = (d > 0) ? ((float)cs[n] / (float)d) : 0.0f;
}

// ---------------- per-hop: candidate edge scoring (WMMA gram) ----------------

__global__ void k_redinit(float* __restrict__ red) {
  unsigned int* u = (unsigned int*)red;
  u[0] = 0x7F800000u;  // +inf  (min over non-negative floats: uint order == float order)
  u[1] = 0u;           // 0.0f  (max)
  red[2] = 0.0f;       // sum
}

// One wave handles 16 candidate edges. Gram terms via V_WMMA_F32_16X16X4_F32:
// A tile (16x4 f32): lane L holds row M=L%16, VGPR v holds K = 2*(L/16)+v.
// B tile (4x16 f32): lane L holds col N=L%16, VGPR v holds K = 2*(L/16)+v.
// => identical per-lane data for X (as A) and X^T (as B): one v2f load serves both.
__global__ __launch_bounds__(128) void k_score(const float* __restrict__ xcur,
                                               const int* __restrict__ ci,
                                               const int* __restrict__ cj,
                                               const float* __restrict__ hom,
                                               float* __restrict__ T,
                                               float* __restrict__ red, int EC) {
  int lane = threadIdx.x & 31;
  long long wave = ((long long)blockIdx.x * blockDim.x + threadIdx.x) >> 5;
  long long base = wave * 16;
  if (base >= EC) return;  // uniform per wave -> EXEC stays all-ones for WMMA
  int r = lane & 15;
  long long eidx = base + r;
  if (eidx >= EC) eidx = EC - 1;  // clamp (duplicate work, writes masked later)
  int ni = ci[eidx];
  int nj = cj[eidx];

  float aii, ajj, aij;
#if __has_builtin(__builtin_amdgcn_wmma_f32_16x16x4_f32)
  {
    int kOff = (lane >> 4) * 2;  // feature sub-offset per lane half
    const float* xi = xcur + (size_t)ni * F_DIM + kOff;
    const float* xj = xcur + (size_t)nj * F_DIM + kOff;
    v8f cii = {}, cjj = {}, cij = {};
#pragma unroll
    for (int k = 0; k < F_DIM / 4; ++k) {
      v2f a = *(const v2f*)(xi + k * 4);
      v2f b = *(const v2f*)(xj + k * 4);
      cii = __builtin_amdgcn_wmma_f32_16x16x4_f32(false, a, false, a, (short)0, cii, false, false);
      cjj = __builtin_amdgcn_wmma_f32_16x16x4_f32(false, b, false, b, (short)0, cjj, false, false);
      cij = __builtin_amdgcn_wmma_f32_16x16x4_f32(false, a, false, b, (short)0, cij, false, false);
    }
    // Diagonal (e,e): held by lane e (v=e) for e<8, lane e+16 (v=e-8) for e>=8.
    int v = lane & 7;
    float dii = pick8(cii, v);
    float djj = pick8(cjj, v);
    float dij = pick8(cij, v);
    int sl = (r < 8) ? r : (r + 16);
    aii = __shfl(dii, sl);
    ajj = __shfl(djj, sl);
    aij = __shfl(dij, sl);
  }
#else
  {
    // scalar fallback: lane halves split the 64 features, combine via xor-16
    const float* xi = xcur + (size_t)ni * F_DIM + (lane >> 4) * 32;
    const float* xj = xcur + (size_t)nj * F_DIM + (lane >> 4) * 32;
    aii = 0.f; ajj = 0.f; aij = 0.f;
#pragma unroll
    for (int f = 0; f < 32; f += 4) {
      v4f a = *(const v4f*)(xi + f);
      v4f b = *(const v4f*)(xj + f);
#pragma unroll
      for (int q = 0; q < 4; ++q) {
        aii = fmaf(a[q], a[q], aii);
        ajj = fmaf(b[q], b[q], ajj);
        aij = fmaf(a[q], b[q], aij);
      }
    }
    aii += __shfl_xor(aii, 16);
    ajj += __shfl_xor(ajj, 16);
    aij += __shfl_xor(aij, 16);
  }
#endif

  // T = sqrt(max(|D||M| - (D.M)^2, 0) + 1e-12), D = xi-xj, M = hdiff*xj
  float h = fabsf(hom[ni] - hom[nj]);
  float dot = h * (aij - ajj);
  float normD = sqrtf(fmaxf(aii - 2.0f * aij + ajj, 0.0f));
  float normM = h * sqrtf(fmaxf(ajj, 0.0f));
  float t = sqrtf(fmaxf(normD * normM - dot * dot, 0.0f) + 1e-12f);

  bool valid = (lane < 16) && (base + lane < EC);
  if (valid) T[base + lane] = t;

  // wave reduction -> global min/max/sum (non-negative floats: uint atomics)
  float tmin = valid ? t : __int_as_float(0x7F800000);
  float tmax = valid ? t : 0.0f;
  float tsum = valid ? t : 0.0f;
#pragma unroll
  for (int o = 16; o > 0; o >>= 1) {
    tmin = fminf(tmin, __shfl_xor(tmin, o));
    tmax = fmaxf(tmax, __shfl_xor(tmax, o));
    tsum += __shfl_xor(tsum, o);
  }
  if (lane == 0) {
    atomicMin((unsigned int*)red + 0, __float_as_uint(tmin));
    atomicMax((unsigned int*)red + 1, __float_as_uint(tmax));
    atomAddF(&red[2], tsum);
  }
}

__global__ void k_scalars(float* __restrict__ red, int EC) {
  unsigned int* u = (unsigned int*)red;
  float tmin = __uint_as_float(u[0]);
  float tmax = __uint_as_float(u[1]);
  float inv = 1.0f / (tmax - tmin + EPS_REF);
  float meanT = red[2] / (float)EC;
  float meanp = 1.0f - (meanT - tmin) * inv;
  float factor = SAMPLING_RATE / (meanp + EPS_REF);
  red[3] = tmin;
  red[4] = inv;
  red[5] = factor;
}

__global__ void k_soft(const float* __restrict__ T, const float* __restrict__ red,
                       float* __restrict__ soft, int EC, int hop) {
  int e = blockIdx.x * blockDim.x + threadIdx.x;
  if (e >= EC) return;
  float tmin = red[3], inv = red[4], factor = red[5];
  float p = 1.0f - (T[e] - tmin) * inv;
  p = fminf(p * factor, 1.0f);
  float l0 = __logf(1.0f - p + EPS_REF);
  float l1 = __logf(p + EPS_REF);
  uint64_t s = (((uint64_t)(hop + 1)) << 33) | ((uint64_t)e << 1);
  float u0 = fmaxf(splitmix_u01(s), 1e-20f);
  float u1 = fmaxf(splitmix_u01(s | 1ULL), 1e-20f);
  float g0 = -__logf(-__logf(u0));
  float g1 = -__logf(-__logf(u1));
  float z = ((l1 + g1) - (l0 + g0)) * TAU_INV;
  soft[e] = 1.0f / (1.0f + __expf(-z));  // softmax over 2 logits -> sigmoid
}

// ---------------- per-hop: normalization + propagation ----------------

__global__ void k_deginit(float* __restrict__ deg, int N) {
  int n = blockIdx.x * blockDim.x + threadIdx.x;
  if (n >= N) return;
  deg[n] = 1.0f;  // self-loop contribution
}

__global__ void k_degacc(const int* __restrict__ src, const int* __restrict__ dst,
                         const int* __restrict__ ci, const float* __restrict__ soft,
                         float* __restrict__ deg, int E, int EC) {
  long long tid = (long long)blockIdx.x * blockDim.x + threadIdx.x;
  if (tid < E) {
    int s = src[tid], d = dst[tid];
    if (s != d) atomAddF(&deg[s], 1.0f);
  } else if (tid < (long long)E + EC) {
    long long e = tid - E;
    atomAddF(&deg[ci[e]], soft[e]);
  }
}

__global__ void k_dinv(const float* __restrict__ deg, float* __restrict__ dinv, int N) {
  int n = blockIdx.x * blockDim.x + threadIdx.x;
  if (n >= N) return;
  dinv[n] = rsqrtf(deg[n] + EPS_REF);
}

__global__ void k_diag(const float* __restrict__ xc, const float* __restrict__ dinv,
                       float* __restrict__ xn, long long total4) {
  long long i = (long long)blockIdx.x * blockDim.x + threadIdx.x;
  if (i >= total4) return;
  int n = (int)(i >> 4);  // 16 float4 per node
  float d = dinv[n];
  float w = d * d;
  v4f v = *(const v4f*)(xc + i * 4);
  v4f o = {w * v[0], w * v[1], w * v[2], w * v[3]};
  *(v4f*)(xn + i * 4) = o;  // self-loop term initializes x_next
}

// 16 threads per edge, 4 features each: one b128 load + 4 hw f32 atomics.
__global__ void k_spmm(const int* __restrict__ src, const int* __restrict__ dst,
                       const int* __restrict__ ci, const int* __restrict__ cj,
                       const float* __restrict__ soft, const float* __restrict__ dinv,
                       const float* __restrict__ xc, float* __restrict__ xn,
                       int E, int EC) {
  long long tid = (long long)blockIdx.x * blockDim.x + threadIdx.x;
  long long tot = ((long long)E + EC) << 4;
  if (tid >= tot) return;
  int f = (int)(tid & 15) * 4;
  long long e = tid >> 4;
  int r, c;
  float w;
  if (e < E) {
    r = src[e];
    c = dst[e];
    if (r == c) return;  // nonself
    w = dinv[r] * dinv[c];
  } else {
    long long e2 = e - E;
    r = ci[e2];
    c = cj[e2];
    w = dinv[r] * soft[e2] * dinv[c];
  }
  v4f v = *(const v4f*)(xc + (size_t)c * F_DIM + f);
  float* o = xn + (size_t)r * F_DIM + f;
  atomAddF(o + 0, w * v[0]);
  atomAddF(o + 1, w * v[1]);
  atomAddF(o + 2, w * v[2]);
  atomAddF(o + 3, w * v[3]);
}

__global__ void k_hid(const float* __restrict__ xn, const float* __restrict__ temp,
                      float* __restrict__ hidden, long long total4, int hop) {
  long long i = (long long)blockIdx.x * blockDim.x + threadIdx.x;
  if (i >= total4) return;
  float t = temp[hop + 1];
  v4f v = *(const v4f*)(xn + i * 4);
  v4f h = *(const v4f*)(hidden + i * 4);
  h[0] += t * v[0];
  h[1] += t * v[1];
  h[2] += t * v[2];
  h[3] += t * v[3];
  *(v4f*)(hidden + i * 4) = h;
}

// ---------------- launch ----------------

extern "C" void kernel_launch(void* const* d_in, const int* in_sizes, int n_in,
                              void* d_out, int out_size, void* d_ws, size_t ws_size,
                              hipStream_t stream) {
  (void)n_in; (void)out_size; (void)ws_size;
  const float* x = (const float*)d_in[0];
  const int* ei = (const int*)d_in[1];     // [2,E] row-major
  const float* cls = (const float*)d_in[2];
  const int* cand = (const int*)d_in[3];   // [2,EC] row-major
  const float* temp = (const float*)d_in[4];

  int N = in_sizes[0] / F_DIM;
  int E = in_sizes[1] / 2;
  int C = in_sizes[2] / N;
  int EC = in_sizes[3] / 2;
  int HOPS = in_sizes[4] - 1;

  const int* src = ei;
  const int* dst = ei + E;
  const int* ci = cand;
  const int* cj = cand + EC;

  // carve workspace
  char* w = (char*)d_ws;
  size_t off = 0;
  auto carve = [&](size_t bytes) -> void* {
    void* p = (void*)(w + off);
    off += (bytes + 255) & ~(size_t)255;
    return p;
  };
  int* pred = (int*)carve((size_t)N * 4);
  int* cs = (int*)carve((size_t)N * 4);
  int* cd = (int*)carve((size_t)N * 4);
  float* hom = (float*)carve((size_t)N * 4);
  float* T = (float*)carve((size_t)EC * 4);
  float* soft = (float*)carve((size_t)EC * 4);
  float* deg = (float*)carve((size_t)N * 4);
  float* dinv = (float*)carve((size_t)N * 4);
  float* xA = (float*)carve((size_t)N * F_DIM * 4);
  float* xB = (float*)carve((size_t)N * F_DIM * 4);
  float* red = (float*)carve(256);

  const int TB = 256;
  long long NF4 = (long long)N * F_DIM / 4;

  k_pred<<<(N + TB - 1) / TB, TB, 0, stream>>>(cls, pred, cs, cd, N, C);
  k_init_x<<<(unsigned)((NF4 + TB - 1) / TB), TB, 0, stream>>>(x, temp, xA, (float*)d_out, NF4);
  k_homcnt<<<(E + TB - 1) / TB, TB, 0, stream>>>(src, dst, pred, cs, cd, E);
  k_homfin<<<(N + TB - 1) / TB, TB, 0, stream>>>(cs, cd, hom, N);

  float* xc = xA;
  float* xn = xB;
  for (int k = 0; k < HOPS; ++k) {
    k_redinit<<<1, 1, 0, stream>>>(red);
    long long waves = ((long long)EC + 15) / 16;
    long long sth = waves * 32;
    k_score<<<(unsigned)((sth + 127) / 128), 128, 0, stream>>>(xc, ci, cj, hom, T, red, EC);
    k_scalars<<<1, 1, 0, stream>>>(red, EC);
    k_soft<<<(EC + TB - 1) / TB, TB, 0, stream>>>(T, red, soft, EC, k);
    k_deginit<<<(N + TB - 1) / TB, TB, 0, stream>>>(deg, N);
    long long dth = (long long)E + EC;
    k_degacc<<<(unsigned)((dth + TB - 1) / TB), TB, 0, stream>>>(src, dst, ci, soft, deg, E, EC);
    k_dinv<<<(N + TB - 1) / TB, TB, 0, stream>>>(deg, dinv, N);
    k_diag<<<(unsigned)((NF4 + TB - 1) / TB), TB, 0, stream>>>(xc, dinv, xn, NF4);
    long long sp = ((long long)E + EC) << 4;
    k_spmm<<<(unsigned)((sp + TB - 1) / TB), TB, 0, stream>>>(src, dst, ci, cj, soft, dinv, xc, xn, E, EC);
    k_hid<<<(unsigned)((NF4 + TB - 1) / TB), TB, 0, stream>>>(xn, temp, (float*)d_out, NF4, k);
    float* t2 = xc; xc = xn; xn = t2;
  }
}